// Loss_23536420782685
// MI455X (gfx1250) — compile-verified
//
#include <hip/hip_runtime.h>
#include <hip/hip_bf16.h>
#include <stdint.h>

// ---- problem constants (from reference) ----
#define SS        7
#define NBOX      2
#define NCLS      20
#define CELL_F    30          // 5*B + C floats per cell (both arrays)
#define L_COORD   5.0f
#define L_NOOBJ   0.5f
#define IMGF      448.0f
#define GRIDF     64.0f
#define EPSF      1e-8f

#define CELLS_PER_TILE 256
#define NTILES         3136            // 16384*49 / 256
#define TILE_FLOATS    (CELLS_PER_TILE * CELL_F)   // 7680 per array
#define TILE_B128      (TILE_FLOATS / 4)           // 1920
#define BOTH_B128      (2 * TILE_B128)             // 3840 = 15 * 256

typedef __attribute__((ext_vector_type(2))) float v2f;
typedef __attribute__((ext_vector_type(8))) float v8f;

__device__ __forceinline__ float iou_one(float bx, float by, float bw, float bh,
                                         float g0, float g1, float g2, float g3,
                                         float garea, float cxg, float cyg) {
    float px = truncf(cxg + bx * GRIDF);
    float py = truncf(cyg + by * GRIDF);
    float pw = truncf(bw * IMGF);
    float ph = truncf(bh * IMGF);
    float x1 = fmaxf(0.0f, px - pw * 0.5f);
    float y1 = fmaxf(0.0f, py - ph * 0.5f);
    float x2 = fminf(IMGF - 1.0f, px + pw * 0.5f);
    float y2 = fminf(IMGF - 1.0f, py + ph * 0.5f);
    float parea = (x2 - x1) * (y2 - y1);
    float lx = fmaxf(x1, g0);
    float rx = fminf(x2, g2);
    float uy = fmaxf(y1, g1);
    float dy = fminf(y2, g3);
    float inter = (rx - lx) * (dy - uy);
    bool  valid = (rx >= lx) && (dy >= uy);
    float denom = parea + garea - inter;
    denom = (denom == 0.0f) ? 1.0f : denom;
    return valid ? (inter / denom) : 0.0f;
}

// One block = one 256-cell tile. Stage bb+gt tile into LDS with async b128
// copies (coalesced), then each thread evaluates its cell from LDS and the
// block tree-reduces 6 partial sums (deterministic, no atomics).
__global__ void __launch_bounds__(256)
yolo_loss_partials(const float* __restrict__ bb, const float* __restrict__ gt,
                   float* __restrict__ ws) {
    __shared__ float sm[2 * TILE_FLOATS];   // [0,7680): bb tile, [7680,15360): gt tile
    __shared__ float red[8 * 6];

    const int tid  = threadIdx.x;
    const int tile = blockIdx.x;

    // ---- async stage: 15 fully-populated b128 loads per thread ----
    const uint64_t bbBase = (uint64_t)(uintptr_t)bb + (uint64_t)tile * (TILE_FLOATS * 4);
    const uint64_t gtBase = (uint64_t)(uintptr_t)gt + (uint64_t)tile * (TILE_FLOATS * 4);
    const uint32_t smBase = (uint32_t)(uintptr_t)sm;

    #pragma unroll
    for (int it = 0; it < 15; ++it) {
        int i = it * 256 + tid;                 // 0..3839
        bool isbb = (i < TILE_B128);
        uint64_t ga = isbb ? (bbBase + (uint64_t)i * 16u)
                           : (gtBase + (uint64_t)(i - TILE_B128) * 16u);
        uint32_t la = smBase + (uint32_t)i * 16u;
        asm volatile("global_load_async_to_lds_b128 %0, %1, off"
                     :: "v"(la), "v"(ga) : "memory");
    }
    asm volatile("s_wait_asynccnt 0" ::: "memory");
    __syncthreads();

    // ---- per-cell loss from LDS ----
    const float* bbC = &sm[tid * CELL_F];
    const float* gtC = &sm[TILE_FLOATS + tid * CELL_F];

    const int cell = tile * CELLS_PER_TILE + tid;   // linear over (N,S,S)
    const int xx = cell % SS;
    const int yy = (cell / SS) % SS;
    const float cxg = (float)xx * GRIDF;
    const float cyg = (float)yy * GRIDF;

    const float g0 = gtC[5], g1 = gtC[6], g2 = gtC[7], g3 = gtC[8];
    const float garea = (g2 - g0) * (g3 - g1);

    const float iou1 = iou_one(bbC[0], bbC[1], bbC[2], bbC[3], g0, g1, g2, g3, garea, cxg, cyg);
    const float iou2 = iou_one(bbC[5], bbC[6], bbC[7], bbC[8], g0, g1, g2, g3, garea, cxg, cyg);

    const bool obj   = (rintf(gtC[9]) != 0.0f);     // jnp.round = half-to-even
    const bool pick1 = (iou1 > iou2);

    const float pb0 = pick1 ? bbC[0] : bbC[5];
    const float pb1 = pick1 ? bbC[1] : bbC[6];
    const float pb2 = pick1 ? bbC[2] : bbC[7];
    const float pb3 = pick1 ? bbC[3] : bbC[8];
    const float pb4 = pick1 ? bbC[4] : bbC[9];
    const float np4 = pick1 ? bbC[9] : bbC[4];
    const float ious = pick1 ? iou1 : iou2;

    const float d0 = gtC[0] - pb0;
    const float d1 = gtC[1] - pb1;
    const float s2 = sqrtf(gtC[2] + EPSF) - sqrtf(pb2 + EPSF);
    const float s3 = sqrtf(gtC[3] + EPSF) - sqrtf(pb3 + EPSF);
    const float coord = L_COORD * (d0 * d0 + d1 * d1 + s2 * s2 + s3 * s3);

    float confo = pb4 - ious;
    confo *= confo;

    float cls = 0.0f;
    #pragma unroll
    for (int c = 0; c < NCLS; ++c) {
        float e = gtC[10 + c] - bbC[10 + c];
        cls += e * e;
    }
    cls /= (float)NCLS;

    const float b1c = bbC[4], b2c = bbC[9];
    const float noobj_cell = L_NOOBJ * b1c * b1c + b2c * b2c;  // note: 0.5 only on b1 (as in ref)
    const float obj_cell   = coord + confo + cls + L_NOOBJ * np4 * np4;

    float acc[6];
    acc[0] = obj ? obj_cell : noobj_cell;
    acc[1] = obj ? coord : 0.0f;
    acc[2] = obj ? (confo + np4 * np4) : noobj_cell;
    acc[3] = obj ? cls : 0.0f;
    acc[4] = obj ? ious : 0.0f;
    acc[5] = obj ? 1.0f : 0.0f;

    // ---- deterministic block reduction: wave32 shfl tree, then 8 leaders ----
    const int lane = tid & 31;
    const int wv   = tid >> 5;
    #pragma unroll
    for (int k = 0; k < 6; ++k) {
        float v = acc[k];
        #pragma unroll
        for (int o = 16; o > 0; o >>= 1)
            v += __shfl_down(v, o, 32);
        if (lane == 0) red[wv * 6 + k] = v;
    }
    __syncthreads();
    if (tid == 0) {
        #pragma unroll
        for (int k = 0; k < 6; ++k) {
            float s = 0.0f;
            #pragma unroll
            for (int w = 0; w < 8; ++w) s += red[w * 6 + k];
            ws[k * NTILES + blockIdx.x] = s;
        }
    }
}

// Final reduction of 6 x 3136 partials with V_WMMA_F32_16X16X4_F32:
// A = 16x4 chunk of partials, B = ones -> D rows accumulate row-sums in f32.
// 3136 = 49 * 64, so 49 WMMAs per output. Column 0 of D lives in lane 0
// (M=0..7 in VGPR0..7) and lane 16 (M=8..15).
__global__ void __launch_bounds__(32)
yolo_loss_finalize(const float* __restrict__ ws, float* __restrict__ out) {
    const int lane = threadIdx.x;          // exactly one wave32, EXEC all ones
    const int m  = lane & 15;
    const int kh = lane >> 4;

    for (int j = 0; j < 6; ++j) {
        const float* P = ws + j * NTILES;
        v8f c = {};
        v2f b;
        b.x = 1.0f; b.y = 1.0f;
        for (int chunk = 0; chunk < 49; ++chunk) {
            const float* q = P + chunk * 64 + m * 4 + kh * 2;
            v2f a;
            a.x = q[0];
            a.y = q[1];
            c = __builtin_amdgcn_wmma_f32_16x16x4_f32(
                    false, a, false, b, (short)0, c, false, false);
        }
        float s = c[0] + c[1] + c[2] + c[3] + c[4] + c[5] + c[6] + c[7];
        float tot = __shfl(s, 0, 32) + __shfl(s, 16, 32);
        if (lane == 0) out[j] = tot;
    }
}

extern "C" void kernel_launch(void* const* d_in, const int* in_sizes, int n_in,
                              void* d_out, int out_size, void* d_ws, size_t ws_size,
                              hipStream_t stream) {
    (void)in_sizes; (void)n_in; (void)out_size; (void)ws_size;
    const float* bb = (const float*)d_in[0];   // bounding_boxes (N,7,7,30)
    const float* gt = (const float*)d_in[1];   // ground_truth  (N,7,7,30)
    float* ws  = (float*)d_ws;                 // 6 * 3136 floats of partials
    float* out = (float*)d_out;                // 6 scalars

    yolo_loss_partials<<<NTILES, 256, 0, stream>>>(bb, gt, ws);
    yolo_loss_finalize<<<1, 32, 0, stream>>>(ws, out);
}